// GCNbaseline_6313601925380
// MI455X (gfx1250) — compile-verified
//
#include <hip/hip_runtime.h>
#include <math.h>

typedef float v2f __attribute__((ext_vector_type(2)));
typedef float v8f __attribute__((ext_vector_type(8)));

#define NN 100000
#define NE 6400000

// ---------------------------------------------------------------------------
// float4 fill (all buffers we clear have element counts divisible by 4)
// ---------------------------------------------------------------------------
__global__ void fill4_kernel(float4* __restrict__ p, float v, int n4) {
    int i = blockIdx.x * blockDim.x + threadIdx.x;
    if (i < n4) p[i] = make_float4(v, v, v, v);
}

// ---------------------------------------------------------------------------
// Degree accumulation: deg[col[e]] += 1  (deg pre-filled with 1.0 = self loop)
// Two edges per thread, vector index load.
// ---------------------------------------------------------------------------
__global__ void deg_accum_kernel(const long long* __restrict__ col,
                                 float* __restrict__ deg, int ne) {
    int e0 = (blockIdx.x * blockDim.x + threadIdx.x) * 2;
    if (e0 >= ne) return;
    longlong2 cc = *(const longlong2*)(col + e0);
    atomicAdd(&deg[(int)cc.x], 1.0f);
    atomicAdd(&deg[(int)cc.y], 1.0f);
}

__global__ void rsqrt_kernel(float* __restrict__ d, int n) {
    int i = blockIdx.x * blockDim.x + threadIdx.x;
    if (i < n) d[i] = rsqrtf(d[i]);  // deg >= 1 always (self loop)
}

// ---------------------------------------------------------------------------
// Dense linear via V_WMMA_F32_16X16X4_F32, K and NOUT compile-time.
// out[m][n] = sum_k in[m][k] * W[n][k] (+ bias[n]); one 16-row tile per wave.
// Weights live in a zero-padded LDS tile -> unconditional ds_load_b64 B fetch.
// Requires nrows % 16 == 0 (NN = 100000 = 6250*16).
//
// A (16x4 f32, ISA layout): lanes 0-15: M=lane, VGPRs {K0,K1};
//                           lanes 16-31: M=lane-16, VGPRs {K2,K3}.
// C/D (16x16 f32, 8 VGPRs): lane half -> M in {0..7} / {8..15}, N = lane%16.
// ---------------------------------------------------------------------------
template <int K, int NOUT, bool BIAS>
__global__ void lin_wmma_kernel(const float* __restrict__ in,
                                const float* __restrict__ W,
                                const float* __restrict__ bias,
                                float* __restrict__ out, int nrows) {
    constexpr int NCH = (K + 3) / 4;
    constexpr int KP  = NCH * 4;
    __shared__ float Wpad[16 * KP];

    // cooperative zero-padded copy of W into LDS (rows >= NOUT and k >= K -> 0)
    for (int idx = threadIdx.x; idx < 16 * KP; idx += blockDim.x) {
        int r = idx / KP, k = idx % KP;
        Wpad[idx] = (r < NOUT && k < K) ? W[r * K + k] : 0.0f;
    }
    __syncthreads();

    int wave = (blockIdx.x * blockDim.x + threadIdx.x) >> 5;
    int lane = threadIdx.x & 31;
    int half = lane >> 4;
    int l    = lane & 15;
    int m0   = wave * 16;

    if (m0 < nrows) {  // wave-uniform; EXEC is all-1s inside for every WMMA
        v8f c = {0.f, 0.f, 0.f, 0.f, 0.f, 0.f, 0.f, 0.f};
        const float* arow = in + (size_t)(m0 + l) * K;
#pragma unroll
        for (int ch = 0; ch < NCH; ++ch) {
            int k0 = ch * 4 + 2 * half;
            v2f a;
            if (ch * 4 + 4 <= K) {          // folded at compile time per iter
                a = *(const v2f*)(arow + k0);
            } else {                         // tail chunk (K%4 != 0)
                a.x = (k0     < K) ? arow[k0]     : 0.0f;
                a.y = (k0 + 1 < K) ? arow[k0 + 1] : 0.0f;
            }
            v2f b = *(const v2f*)(&Wpad[l * KP + k0]);
            // 8 args: (neg_a, A, neg_b, B, c_mod, C, reuse_a, reuse_b)
            c = __builtin_amdgcn_wmma_f32_16x16x4_f32(
                false, a, false, b, (short)0, c, false, false);
        }
        if (l < NOUT) {
            float bv = BIAS ? bias[l] : 0.0f;
#pragma unroll
            for (int v = 0; v < 8; ++v) {
                int m = m0 + 8 * half + v;
                out[m * NOUT + l] = c[v] + bv;
            }
        }
    }
}

// ---------------------------------------------------------------------------
// Edge scatter: agg[col] += lin[row] * dis[row] * dis[col]
// Two edges per thread; b128 index loads; vector lin loads; f32 atomics (L2).
// ---------------------------------------------------------------------------
template <int F>
__global__ void scatter_kernel(const long long* __restrict__ ei,
                               const float* __restrict__ lin,
                               const float* __restrict__ dis,
                               float* __restrict__ agg, int ne) {
    int e0 = (blockIdx.x * blockDim.x + threadIdx.x) * 2;
    if (e0 >= ne) return;
    longlong2 rr = *(const longlong2*)(ei + e0);        // sources
    longlong2 cc = *(const longlong2*)(ei + ne + e0);   // targets
#pragma unroll
    for (int j = 0; j < 2; ++j) {
        int r = (int)(j ? rr.y : rr.x);
        int c = (int)(j ? cc.y : cc.x);
        float nrm = dis[r] * dis[c];
        if (F == 4) {
            float4 v = *(const float4*)(lin + r * 4);
            atomicAdd(&agg[c * 4 + 0], v.x * nrm);
            atomicAdd(&agg[c * 4 + 1], v.y * nrm);
            atomicAdd(&agg[c * 4 + 2], v.z * nrm);
            atomicAdd(&agg[c * 4 + 3], v.w * nrm);
        } else {
            float2 v = *(const float2*)(lin + r * 2);
            atomicAdd(&agg[c * 2 + 0], v.x * nrm);
            atomicAdd(&agg[c * 2 + 1], v.y * nrm);
        }
    }
}

// ---------------------------------------------------------------------------
// Finalize: h[i] = act(agg[i] + dis[i]^2 * lin[i] + b)  (self loop folded in)
// ---------------------------------------------------------------------------
template <int F, bool TANH>
__global__ void finalize_kernel(const float* __restrict__ agg,
                                const float* __restrict__ lin,
                                const float* __restrict__ dis,
                                const float* __restrict__ b,
                                float* __restrict__ out, int n) {
    int i = blockIdx.x * blockDim.x + threadIdx.x;
    if (i >= n) return;
    float d2 = dis[i] * dis[i];
    if (F == 4) {
        float4 av = *(const float4*)(agg + i * 4);
        float4 lv = *(const float4*)(lin + i * 4);
        float4 o;
        o.x = av.x + d2 * lv.x + b[0];
        o.y = av.y + d2 * lv.y + b[1];
        o.z = av.z + d2 * lv.z + b[2];
        o.w = av.w + d2 * lv.w + b[3];
        if (TANH) { o.x = tanhf(o.x); o.y = tanhf(o.y);
                    o.z = tanhf(o.z); o.w = tanhf(o.w); }
        *(float4*)(out + i * 4) = o;
    } else {
        float2 av = *(const float2*)(agg + i * 2);
        float2 lv = *(const float2*)(lin + i * 2);
        float2 o;
        o.x = av.x + d2 * lv.x + b[0];
        o.y = av.y + d2 * lv.y + b[1];
        if (TANH) { o.x = tanhf(o.x); o.y = tanhf(o.y); }
        *(float2*)(out + i * 2) = o;
    }
}

// ---------------------------------------------------------------------------
extern "C" void kernel_launch(void* const* d_in, const int* in_sizes, int n_in,
                              void* d_out, int out_size, void* d_ws, size_t ws_size,
                              hipStream_t stream) {
    const float*     x  = (const float*)d_in[0];
    const long long* ei = (const long long*)d_in[1];  // [2, NE] int64
    const float* W1 = (const float*)d_in[2];
    const float* b1 = (const float*)d_in[3];
    const float* W2 = (const float*)d_in[4];
    const float* b2 = (const float*)d_in[5];
    const float* W3 = (const float*)d_in[6];
    const float* b3 = (const float*)d_in[7];
    const float* Wc = (const float*)d_in[8];
    const float* bc = (const float*)d_in[9];

    float* outp = (float*)d_out;            // [NN,4]
    float* h1   = outp + NN * 4;            // [NN,4]
    float* h2   = outp + NN * 8;            // [NN,4]
    float* h3   = outp + NN * 12;           // [NN,2]

    float* ws  = (float*)d_ws;
    float* dis = ws;                        // NN        (deg, then deg^-1/2)
    float* lin = ws + NN;                   // NN*4
    float* agg = ws + NN * 5;               // NN*4      total 3.6 MB

    const int TPB = 256;
    const int pair_blocks = (NE / 2 + TPB - 1) / TPB;     // 2 edges/thread
    const int ntiles      = NN / 16;                      // one wave per tile
    const int lin_blocks  = (ntiles + (TPB / 32) - 1) / (TPB / 32);
    const int node_blocks = (NN + TPB - 1) / TPB;

    // ---- degree / normalization -----------------------------------------
    fill4_kernel<<<(NN / 4 + TPB - 1) / TPB, TPB, 0, stream>>>((float4*)dis, 1.0f, NN / 4);
    deg_accum_kernel<<<pair_blocks, TPB, 0, stream>>>(ei + NE, dis, NE);
    rsqrt_kernel<<<node_blocks, TPB, 0, stream>>>(dis, NN);

    // ---- layer 1: x[NN,34] -> h1[NN,4] ----------------------------------
    lin_wmma_kernel<34, 4, false><<<lin_blocks, TPB, 0, stream>>>(x, W1, nullptr, lin, NN);
    fill4_kernel<<<(NN + TPB - 1) / TPB, TPB, 0, stream>>>((float4*)agg, 0.0f, NN);
    scatter_kernel<4><<<pair_blocks, TPB, 0, stream>>>(ei, lin, dis, agg, NE);
    finalize_kernel<4, true><<<node_blocks, TPB, 0, stream>>>(agg, lin, dis, b1, h1, NN);

    // ---- layer 2: h1[NN,4] -> h2[NN,4] ----------------------------------
    lin_wmma_kernel<4, 4, false><<<lin_blocks, TPB, 0, stream>>>(h1, W2, nullptr, lin, NN);
    fill4_kernel<<<(NN + TPB - 1) / TPB, TPB, 0, stream>>>((float4*)agg, 0.0f, NN);
    scatter_kernel<4><<<pair_blocks, TPB, 0, stream>>>(ei, lin, dis, agg, NE);
    finalize_kernel<4, true><<<node_blocks, TPB, 0, stream>>>(agg, lin, dis, b2, h2, NN);

    // ---- layer 3: h2[NN,4] -> h3[NN,2] ----------------------------------
    lin_wmma_kernel<4, 2, false><<<lin_blocks, TPB, 0, stream>>>(h2, W3, nullptr, lin, NN);
    fill4_kernel<<<(NN / 2 + TPB - 1) / TPB, TPB, 0, stream>>>((float4*)agg, 0.0f, NN / 2);
    scatter_kernel<2><<<pair_blocks, TPB, 0, stream>>>(ei, lin, dis, agg, NE);
    finalize_kernel<2, true><<<node_blocks, TPB, 0, stream>>>(agg, lin, dis, b3, h3, NN);

    // ---- classifier: out = h3 @ Wc.T + bc (no message passing) ----------
    lin_wmma_kernel<2, 4, true><<<lin_blocks, TPB, 0, stream>>>(h3, Wc, bc, outp, NN);
}